// Model_47751446396960
// MI455X (gfx1250) — compile-verified
//
#include <hip/hip_runtime.h>

#define RES 128
#define PIX (RES * RES)
#define PAD_STRIDE 132          // indices -1..130 (border zeros), interior at +1
#define CENTER 63.5f
#define NB 32
#define NC 16
#define NA 64
#define ANG_PER_BLOCK 4

typedef __attribute__((ext_vector_type(2))) float v2f;
typedef __attribute__((ext_vector_type(8))) float v8f;

// ---------------------------------------------------------------------------
// Pass 1: channel collapse  y[b,h,w] = sum_c w[c] * x[b,c,h,w]
// Pure streaming pass: 33.5 MB read, 2 MB write. float4 vectorized.
// ---------------------------------------------------------------------------
__global__ __launch_bounds__(256) void collapse_kernel(
    const float* __restrict__ x, const float* __restrict__ w,
    float* __restrict__ y)
{
    int idx = blockIdx.x * 256 + threadIdx.x;   // 131072 threads, 4 floats each
    int b   = idx >> 12;                        // 4096 float4 per image
    int p4  = idx & 4095;

    const float4* xb = reinterpret_cast<const float4*>(x) + (size_t)b * NC * (PIX / 4) + p4;
    float a0 = 0.f, a1 = 0.f, a2 = 0.f, a3 = 0.f;
#pragma unroll
    for (int c = 0; c < NC; ++c) {
        float  wc = w[c];                       // uniform -> SMEM load
        float4 v  = xb[c * (PIX / 4)];
        a0 = fmaf(wc, v.x, a0);
        a1 = fmaf(wc, v.y, a1);
        a2 = fmaf(wc, v.z, a2);
        a3 = fmaf(wc, v.w, a3);
    }
    float4* yp = reinterpret_cast<float4*>(y) + (size_t)b * (PIX / 4) + p4;
    *yp = make_float4(a0, a1, a2, a3);
}

// Bilinear sample from the zero-padded LDS tile. Coordinates clamped to
// [-1, 128]: out-of-range samples hit the zero border with the exact same
// result as the reference's validity masks (bilinear is continuous, and the
// partially-valid edge cases interpolate against zeros identically).
__device__ __forceinline__ float sample_bilinear(const float* __restrict__ lds,
                                                 float dp, float jp,
                                                 float c, float s)
{
    float xs = fmaf(dp, c, fmaf(-jp, s, CENTER));
    float ys = fmaf(dp, s, fmaf( jp, c, CENTER));
    xs = fminf(fmaxf(xs, -1.0f), 128.0f);
    ys = fminf(fmaxf(ys, -1.0f), 128.0f);
    float x0f = floorf(xs), y0f = floorf(ys);
    float wx = xs - x0f, wy = ys - y0f;
    int x0 = (int)x0f + 1;                      // shift into padded tile
    int y0 = (int)y0f + 1;
    const float* p = lds + y0 * PAD_STRIDE + x0;
    float v00 = p[0], v01 = p[1];
    float v10 = p[PAD_STRIDE], v11 = p[PAD_STRIDE + 1];
    float top = fmaf(wx, v01 - v00, v00);
    float bot = fmaf(wx, v11 - v10, v10);
    return fmaf(wy, bot - top, top);
}

// ---------------------------------------------------------------------------
// Pass 2: Radon projection of y[b] with WMMA reduction.
// Block = (one b, 4 angles). Image staged into zero-padded 132x132 LDS tile.
// Each wave32 handles (angle, 16-row d-tile) units; the sum over the 128
// sample positions j is accumulated on the matrix pipe via
// V_WMMA_F32_16X16X4_F32 with an all-ones B matrix: D = A*1 + C == row sums.
// ---------------------------------------------------------------------------
__global__ __launch_bounds__(256) void radon_wmma_kernel(
    const float* __restrict__ y, const float* __restrict__ angles,
    const float* __restrict__ bias, float* __restrict__ out)
{
    extern __shared__ float lds[];              // PAD_STRIDE*PAD_STRIDE floats

    const int b   = blockIdx.x / (NA / ANG_PER_BLOCK);
    const int ag  = blockIdx.x % (NA / ANG_PER_BLOCK);
    const int tid = threadIdx.x;

    // Zero the whole padded tile, then fill the interior.
    for (int i = tid; i < PAD_STRIDE * PAD_STRIDE; i += 256) lds[i] = 0.0f;
    __syncthreads();
    const float* yb = y + (size_t)b * PIX;
    for (int i = tid; i < PIX; i += 256) {
        int r = i >> 7, col = i & 127;
        lds[(r + 1) * PAD_STRIDE + (col + 1)] = yb[i];
    }
    __syncthreads();

    const float bval = bias[0];
    const int wave = tid >> 5;
    const int lane = tid & 31;
    const int m    = lane & 15;                 // A-matrix row M within tile
    const int half = lane >> 4;                 // 0: K=0,1   1: K=2,3

    v2f bmat;                                   // all-ones B (4x16)
    bmat[0] = 1.0f;
    bmat[1] = 1.0f;

#pragma unroll
    for (int k = 0; k < 4; ++k) {
        const int u    = wave + 8 * k;          // 32 units = 4 angles x 8 tiles
        const int aL   = u >> 3;
        const int tile = u & 7;
        const int aG   = ag * ANG_PER_BLOCK + aL;

        const float th = angles[aG];
        const float c  = cosf(th);
        const float s  = sinf(th);
        const float dp = (float)(tile * 16 + m) - CENTER;

        v8f acc = {};
        for (int jb = 0; jb < RES; jb += 4) {
            float j0 = (float)(jb + 2 * half) - CENTER;
            v2f amat;
            amat[0] = sample_bilinear(lds, dp, j0,        c, s); // K = 2*half
            amat[1] = sample_bilinear(lds, dp, j0 + 1.0f, c, s); // K = 2*half+1
            acc = __builtin_amdgcn_wmma_f32_16x16x4_f32(
                false, amat, false, bmat, (short)0, acc, false, false);
        }

        // Column N=0 of D lives in lane 0 (M=0..7) and lane 16 (M=8..15),
        // one row per accumulator VGPR -> contiguous d-range stores.
        if (m == 0) {
            float* op = out + ((size_t)b * NA + aG) * RES + tile * 16 + half * 8;
#pragma unroll
            for (int r = 0; r < 8; ++r) op[r] = acc[r] + bval;
        }
    }
}

extern "C" void kernel_launch(void* const* d_in, const int* in_sizes, int n_in,
                              void* d_out, int out_size, void* d_ws, size_t ws_size,
                              hipStream_t stream) {
    const float* x      = (const float*)d_in[0];   // (32,16,128,128)
    const float* angles = (const float*)d_in[1];   // (64,)
    const float* conv_w = (const float*)d_in[2];   // (1,16,1,1)
    const float* conv_b = (const float*)d_in[3];   // (1,)
    float* out = (float*)d_out;                    // (32,1,64,128)
    float* yws = (float*)d_ws;                     // 2 MB collapsed images

    collapse_kernel<<<(NB * PIX / 4) / 256, 256, 0, stream>>>(x, conv_w, yws);

    radon_wmma_kernel<<<NB * (NA / ANG_PER_BLOCK), 256,
                        PAD_STRIDE * PAD_STRIDE * sizeof(float), stream>>>(
        yws, angles, conv_b, out);
}